// Quantization_14628658610753
// MI455X (gfx1250) — compile-verified
//
#include <hip/hip_runtime.h>
#include <hip/hip_bf16.h>

// ---------------------------------------------------------------------------
// Problem constants (from reference setup_inputs)
#define QN   4096          // rows (vectors)
#define QE   768           // embedding dim
#define QP   1024          // num_pq centroids
#define QK   128           // num_codebook
#define QC   256           // num_codeword
#define QS   6             // subvec size
#define QTEMP_LOG2E 144.2695040888963f   // (1/TEMP) * log2(e) = 100 * log2(e)

typedef __attribute__((ext_vector_type(2)))  float   v2f;
typedef __attribute__((ext_vector_type(8)))  float   v8f;

__device__ __forceinline__ v2f qz_ld2(const float* p) {
  const float2 t = *(const float2*)p;   // 8B-aligned (even float offset)
  v2f r; r.x = t.x; r.y = t.y;
  return r;
}

// ---------------------------------------------------------------------------
// Kernel 0: row squared norms.  blocks [0,4096) -> vectors, [4096,5120) -> pq
__global__ __launch_bounds__(256)
void qz_row_norms(const float* __restrict__ vec,
                  const float* __restrict__ pq,
                  float* __restrict__ ws) {
  const int row = blockIdx.x;
  const int tid = threadIdx.x;
  const float* src = (row < QN) ? (vec + (size_t)row * QE)
                                : (pq  + (size_t)(row - QN) * QE);
  __shared__ float sv[256];
  float s = 0.f;
  for (int i = tid; i < QE; i += 256) { const float v = src[i]; s += v * v; }
  sv[tid] = s;
  __syncthreads();
  for (int off = 128; off > 0; off >>= 1) {
    if (tid < off) sv[tid] += sv[tid + off];
    __syncthreads();
  }
  if (tid == 0) ws[row] = sv[0];
}

// ---------------------------------------------------------------------------
// Kernel 1: pq_simi = -(n1 + n2 - 2 * V @ Aq^T) via f32 WMMA 16x16x4.
// Each wave computes a 16x64 output strip: 4 accumulators share one A
// fragment per K-step (1.25 loads/WMMA instead of 2).  4 waves per block
// cover 16 rows x 256 cols.
__global__ __launch_bounds__(128)
void qz_pq_gemm(const float* __restrict__ A,   // vectors   [4096,768]
                const float* __restrict__ B,   // pq_assign [1024,768]
                const float* __restrict__ n1,  // [4096]
                const float* __restrict__ n2,  // [1024]
                float* __restrict__ simi) {    // [4096,1024]
  const int lane = threadIdx.x & 31;
  const int wave = threadIdx.x >> 5;
  const int half = lane >> 4;     // 0: lanes 0-15, 1: lanes 16-31
  const int l    = lane & 15;
  const int m0   = blockIdx.y * 16;
  const int c0   = (blockIdx.x * 4 + wave) * 64;   // 64-wide strip per wave

  const float* arow = A + (size_t)(m0 + l) * QE;
  const float* b0   = B + (size_t)(c0 +  0 + l) * QE;
  const float* b1   = B + (size_t)(c0 + 16 + l) * QE;
  const float* b2   = B + (size_t)(c0 + 32 + l) * QE;
  const float* b3   = B + (size_t)(c0 + 48 + l) * QE;

  v8f acc0 = {}, acc1 = {}, acc2 = {}, acc3 = {};

  // f32 WMMA 16x16x4.  A frag (ISA 7.12.2): lanes 0-15 hold K=k..k+1 of row
  // M=l, lanes 16-31 hold K=k+2..k+3.  B frags mirror with N=l.
  #pragma unroll 2
  for (int k = 0; k < QE; k += 4) {
    const int ko = k + 2 * half;
    const v2f a  = qz_ld2(arow + ko);
    const v2f q0 = qz_ld2(b0 + ko);
    const v2f q1 = qz_ld2(b1 + ko);
    const v2f q2 = qz_ld2(b2 + ko);
    const v2f q3 = qz_ld2(b3 + ko);
    acc0 = __builtin_amdgcn_wmma_f32_16x16x4_f32(false, a, false, q0,
                                                 (short)0, acc0, false, false);
    acc1 = __builtin_amdgcn_wmma_f32_16x16x4_f32(false, a, false, q1,
                                                 (short)0, acc1, false, false);
    acc2 = __builtin_amdgcn_wmma_f32_16x16x4_f32(false, a, false, q2,
                                                 (short)0, acc2, false, false);
    acc3 = __builtin_amdgcn_wmma_f32_16x16x4_f32(false, a, false, q3,
                                                 (short)0, acc3, false, false);
  }

  // Epilogue: C/D layout -> VGPR r holds M = r + 8*half, N = l.
  v8f accs[4] = {acc0, acc1, acc2, acc3};
  #pragma unroll
  for (int t = 0; t < 4; ++t) {
    const int col = c0 + 16 * t + l;
    const float nn2 = n2[col];
    #pragma unroll
    for (int r = 0; r < 8; ++r) {
      const int row = m0 + r + 8 * half;
      simi[(size_t)row * QP + col] = -(n1[row] + nn2 - 2.0f * accs[t][r]);
    }
  }
}

// ---------------------------------------------------------------------------
// Kernel 2: per-row argmax over 1024 centroids (first-index tie-break) and
// gather of the winning centroid row.  Wave-level shfl reductions.
__global__ __launch_bounds__(256)
void qz_argmax_gather(const float* __restrict__ simi,
                      const float* __restrict__ pq,
                      float* __restrict__ centers) {
  const int n    = blockIdx.x;
  const int tid  = threadIdx.x;
  const int lane = tid & 31;
  const int wave = tid >> 5;
  __shared__ float sv[8];
  __shared__ int   si[8];

  float best = -3.4028235e38f;
  int   bi   = 0;
  for (int j = tid; j < QP; j += 256) {      // ascending j => first-max kept
    const float v = simi[(size_t)n * QP + j];
    if (v > best) { best = v; bi = j; }
  }
  #pragma unroll
  for (int off = 16; off > 0; off >>= 1) {
    const float ov = __shfl_xor(best, off, 32);
    const int   oi = __shfl_xor(bi,   off, 32);
    if (ov > best || (ov == best && oi < bi)) { best = ov; bi = oi; }
  }
  if (lane == 0) { sv[wave] = best; si[wave] = bi; }
  __syncthreads();
  float m = sv[0]; int idx = si[0];
  #pragma unroll
  for (int w = 1; w < 8; ++w) {
    const float v = sv[w]; const int i = si[w];
    if (v > m || (v == m && i < idx)) { m = v; idx = i; }
  }
  const float* src = pq + (size_t)idx * QE;
  for (int i = tid; i < QE; i += 256)
    centers[(size_t)n * QE + i] = src[i];
}

// ---------------------------------------------------------------------------
// Kernel 3: residual code selection + softmax + hard-assign + L2 norm.
// One block (8 waves) per row.  Each WAVE owns whole codebooks (k = wave,
// wave+8, ...); each lane owns 8 contiguous codewords.  All max/sum
// reductions are wave-internal shfl_xor -> zero barriers in the k-loop.
__global__ __launch_bounds__(256)
void qz_code_select(const float* __restrict__ vectors,
                    const float* __restrict__ codebooks,   // [128,256,6]
                    const float* __restrict__ centers,     // [4096,768]
                    float* __restrict__ out_assign,        // [4096,128,256]
                    float* __restrict__ out_qvec) {        // [4096,768]
  const int n    = blockIdx.x;
  const int tid  = threadIdx.x;
  const int lane = tid & 31;
  const int wave = tid >> 5;       // 0..7
  __shared__ float resid[QE];
  __shared__ float qvec[QE];
  __shared__ float sred[8];

  for (int i = tid; i < QE; i += 256) {
    const float pc = centers[(size_t)n * QE + i];
    resid[i] = vectors[(size_t)n * QE + i] - pc;
    qvec[i]  = pc;
  }
  __syncthreads();

  float* arow = out_assign + (size_t)n * (QK * QC);

  for (int k = wave; k < QK; k += 8) {
    float r[QS];
    #pragma unroll
    for (int s = 0; s < QS; ++s) r[s] = resid[k * QS + s];
    float r2 = 0.f;
    #pragma unroll
    for (int s = 0; s < QS; ++s) r2 += r[s] * r[s];

    // 8 codewords per lane: c = lane*8 + j.  48 contiguous floats,
    // 192B-aligned -> 12x float4 loads.
    const float* cb = codebooks + ((size_t)k * QC + lane * 8) * QS;
    float cw[48];
    #pragma unroll
    for (int t = 0; t < 12; ++t) ((float4*)cw)[t] = ((const float4*)cb)[t];

    float p[8];
    float lmax = -3.4028235e38f;
    int   lidx = 0;
    #pragma unroll
    for (int j = 0; j < 8; ++j) {
      float rc = 0.f, cc = 0.f;
      #pragma unroll
      for (int s = 0; s < QS; ++s) {
        const float c = cw[j * QS + s];
        rc += r[s] * c;
        cc += c * c;
      }
      p[j] = -(r2 - 2.f * rc + cc);
      if (p[j] > lmax) { lmax = p[j]; lidx = lane * 8 + j; }  // first-max in lane
    }

    // wave argmax (value, first index)
    float m = lmax; int mi = lidx;
    #pragma unroll
    for (int off = 16; off > 0; off >>= 1) {
      const float om = __shfl_xor(m,  off, 32);
      const int   oi = __shfl_xor(mi, off, 32);
      if (om > m || (om == m && oi < mi)) { m = om; mi = oi; }
    }

    // stable softmax of p/TEMP in log2 domain
    float e[8];
    float ls = 0.f;
    #pragma unroll
    for (int j = 0; j < 8; ++j) {
      e[j] = exp2f((p[j] - m) * QTEMP_LOG2E);
      ls += e[j];
    }
    #pragma unroll
    for (int off = 16; off > 0; off >>= 1) ls += __shfl_xor(ls, off, 32);
    const float inv = 1.0f / ls;

    float* dst = arow + k * QC + lane * 8;    // 8 contiguous floats per lane
    #pragma unroll
    for (int j = 0; j < 8; ++j) dst[j] = e[j] * inv;

    // hard assignment (STE forward value): lane owning argmax adds codeword.
    // Each k is owned by exactly one wave -> no cross-wave conflict.
    if ((mi >> 3) == lane) {
      #pragma unroll
      for (int s = 0; s < QS; ++s) qvec[k * QS + s] += cw[(mi & 7) * QS + s];
    }
  }
  __syncthreads();

  // L2 normalize q = centers + quant
  float ss = 0.f;
  for (int i = tid; i < QE; i += 256) ss += qvec[i] * qvec[i];
  #pragma unroll
  for (int off = 16; off > 0; off >>= 1) ss += __shfl_xor(ss, off, 32);
  if (lane == 0) sred[wave] = ss;
  __syncthreads();
  float tot = 0.f;
  #pragma unroll
  for (int w = 0; w < 8; ++w) tot += sred[w];
  const float scale = 1.0f / fmaxf(sqrtf(tot), 1e-12f);
  for (int i = tid; i < QE; i += 256)
    out_qvec[(size_t)n * QE + i] = qvec[i] * scale;
}

// ---------------------------------------------------------------------------
extern "C" void kernel_launch(void* const* d_in, const int* in_sizes, int n_in,
                              void* d_out, int out_size, void* d_ws, size_t ws_size,
                              hipStream_t stream) {
  (void)in_sizes; (void)n_in; (void)out_size; (void)ws_size;
  const float* vectors   = (const float*)d_in[0];   // [4096,768]
  const float* pq_assign = (const float*)d_in[1];   // [1024,768]
  const float* codebooks = (const float*)d_in[2];   // [128,256,6]

  float* out = (float*)d_out;
  float* simi    = out;                                    // 4096*1024
  float* centers = out + (size_t)QN * QP;                  // 4096*768
  float* assign  = centers + (size_t)QN * QE;              // 4096*128*256
  float* qvecs   = assign + (size_t)QN * QK * QC;          // 4096*768

  float* wsf = (float*)d_ws;
  float* n1 = wsf;            // [4096]
  float* n2 = wsf + QN;       // [1024]

  qz_row_norms<<<QN + QP, 256, 0, stream>>>(vectors, pq_assign, wsf);
  qz_pq_gemm<<<dim3(QP / 256, QN / 16), 128, 0, stream>>>(vectors, pq_assign,
                                                          n1, n2, simi);
  qz_argmax_gather<<<QN, 256, 0, stream>>>(simi, pq_assign, centers);
  qz_code_select<<<QN, 256, 0, stream>>>(vectors, codebooks, centers,
                                         assign, qvecs);
}